// MultiTokenizer_59596966199732
// MI455X (gfx1250) — compile-verified
//
#include <hip/hip_runtime.h>

// ---- problem constants (from reference) ----
constexpr int B_    = 128;
constexpr int T_    = 16384;
constexpr int E_    = 256;
constexpr int MAXW  = 32;
constexpr int SMAX  = 2048;          // T / min(W) = 16384/8
constexpr float TARGET_MM = 0.6f;
constexpr float SOS_MM_S  = 1540000.0f;

typedef __attribute__((ext_vector_type(2))) float v2f;
typedef __attribute__((ext_vector_type(8))) float v8f;

// ---------------------------------------------------------------------------
// Setup kernel: one block per sample.
//  - reduce signal_mask -> length
//  - route branch from sampling frequency
//  - write padding_mask, W_b, ts_us, fs outputs (non-temporal: write-once)
//  - store per-sample meta {W, S_b, branch} into workspace for the GEMM kernel
// ---------------------------------------------------------------------------
__global__ __launch_bounds__(256)
void mt_setup(const unsigned char* __restrict__ mask,
              const float* __restrict__ fs_in,
              float* __restrict__ pm_out,   // (B, SMAX) as 0/1 float
              float* __restrict__ wb_out,   // (B,)
              float* __restrict__ ts_out,   // (B, SMAX)
              float* __restrict__ fs_out,   // (B,)
              int4*  __restrict__ meta)     // (B,) {W, Sb, branch, fsc bits}
{
    const int b   = blockIdx.x;
    const int tid = threadIdx.x;

    __shared__ int   s_len;
    __shared__ int   s_W, s_Sb;
    __shared__ float s_fsc;

    if (tid == 0) s_len = 0;
    __syncthreads();

    // mask bytes are 0/1 -> sum via popcount on dword loads
    const unsigned int* m32 = (const unsigned int*)(mask + (size_t)b * T_);
    int cnt = 0;
    for (int t = tid; t < T_ / 4; t += 256)
        cnt += __popc(m32[t] & 0x01010101u);
    atomicAdd(&s_len, cnt);
    __syncthreads();

    if (tid == 0) {
        const float fs  = fs_in[b];
        const float fsc = fmaxf(fs, 1.0f);
        const int ws[3] = {8, 16, 32};
        int best = 0; float bestd = 3.4e38f;
        for (int i = 0; i < 3; ++i) {
            float depth = (float)ws[i] * SOS_MM_S / (2.0f * fsc);
            float d = fabsf(depth - TARGET_MM);
            if (d < bestd) { bestd = d; best = i; }
        }
        const int W  = ws[best];
        int Sb = s_len / W;
        if (Sb > SMAX) Sb = SMAX;
        s_W = W; s_Sb = Sb; s_fsc = fsc;
        meta[b]   = make_int4(W, Sb, best, __float_as_int(fsc));
        wb_out[b] = (float)W;
        fs_out[b] = fs;
    }
    __syncthreads();

    const int   W   = s_W;
    const int   Sb  = s_Sb;
    const float fsc = s_fsc;
    const float Wf  = (float)W;
    for (int i = tid; i < SMAX; i += 256) {
        __builtin_nontemporal_store((i < Sb) ? 1.0f : 0.0f,
                                    &pm_out[(size_t)b * SMAX + i]);
        __builtin_nontemporal_store(((float)i * Wf + 0.5f * Wf) / fsc * 1.0e6f,
                                    &ts_out[(size_t)b * SMAX + i]);
    }
}

// ---------------------------------------------------------------------------
// GEMM kernel: grid (SMAX/16, B). 256 threads = 8 waves.
// Block handles a 16-row patch tile of one sample; wave w handles embed
// columns [32w, 32w+32) as two 16x16 WMMA tiles.
// tokens[b, s0+m, e] = sum_k sA[m, k] * weight[br, e, k] + bias[br, e]
// using V_WMMA_F32_16X16X4_F32, K-loop of W/4 steps.
// Token stores are non-temporal: 268 MB write-once stream > 192 MB L2;
// keep L2 for the weight tables that every block re-reads.
// ---------------------------------------------------------------------------
__global__ __launch_bounds__(256)
void mt_gemm(const float* __restrict__ signal,
             const float* __restrict__ weight,   // (3, E, MAXW)
             const float* __restrict__ bias,     // (3, E)
             const int4*  __restrict__ meta,
             float* __restrict__ tokens)         // (B, SMAX, E)
{
    const int tile_s = blockIdx.x;          // 0..SMAX/16-1
    const int b      = blockIdx.y;
    const int tid    = threadIdx.x;
    const int wave   = tid >> 5;
    const int lane   = tid & 31;

    const int4 m = meta[b];                 // uniform scalar load
    const int W  = m.x;
    const int Sb = m.y;
    const int br = m.z;

    const int s0    = tile_s * 16;
    const int mrow  = lane & 15;            // A-row / C-column lane index
    const int khalf = lane >> 4;            // 0: k+0/k+1, 1: k+2/k+3
    const int e0    = wave * 32;
    const int eA    = e0 + mrow;
    const int eB    = e0 + 16 + mrow;

    const float biasA = bias[br * E_ + eA];
    const float biasB = bias[br * E_ + eB];
    float* out = tokens + ((size_t)b * SMAX + (size_t)s0) * E_;

    // Whole tile past valid region: tokens = bias only (uniform per block).
    if (s0 >= Sb) {
        #pragma unroll
        for (int r = 0; r < 8; ++r) {
            const int row = r + 8 * khalf;
            __builtin_nontemporal_store(biasA, &out[(size_t)row * E_ + eA]);
            __builtin_nontemporal_store(biasB, &out[(size_t)row * E_ + eB]);
        }
        return;
    }

    // Stage the 16 contiguous patch rows (16*W floats) into LDS, zeroing
    // rows >= Sb so masked rows reduce to bias.
    __shared__ float sA[16 * MAXW];
    const float* sig = signal + (size_t)b * T_ + (size_t)s0 * W;
    const int nA = 16 * W;
    for (int i = tid; i < nA; i += 256) {
        const int row = s0 + i / W;
        sA[i] = (row < Sb) ? sig[i] : 0.0f;
    }
    __syncthreads();

    v8f c0 = {};
    v8f c1 = {};
    const int ksteps = W >> 2;              // uniform: 2, 4 or 8
    for (int kc = 0; kc < ksteps; ++kc) {
        const int k = kc * 4 + 2 * khalf;   // even -> 8B aligned
        // A fragment: A[mrow][k], A[mrow][k+1]
        v2f a = *(const v2f*)(&sA[mrow * W + k]);
        // B fragments: weight[br][e][k], weight[br][e][k+1]
        const float* wp = weight + ((size_t)(br * E_)) * MAXW;
        v2f bA = *(const v2f*)(wp + (size_t)eA * MAXW + k);
        v2f bB = *(const v2f*)(wp + (size_t)eB * MAXW + k);
        c0 = __builtin_amdgcn_wmma_f32_16x16x4_f32(
                 false, a, false, bA, (short)0, c0, false, false);
        c1 = __builtin_amdgcn_wmma_f32_16x16x4_f32(
                 false, a, false, bB, (short)0, c1, false, false);
    }

    // C/D layout: VGPR r -> row = r + 8*(lane/16), col = lane%16.
    // Lanes 0-15 / 16-31 each store a contiguous 64B segment per row.
    #pragma unroll
    for (int r = 0; r < 8; ++r) {
        const int row = r + 8 * khalf;
        __builtin_nontemporal_store(c0[r] + biasA, &out[(size_t)row * E_ + eA]);
        __builtin_nontemporal_store(c1[r] + biasB, &out[(size_t)row * E_ + eB]);
    }
}

// ---------------------------------------------------------------------------
extern "C" void kernel_launch(void* const* d_in, const int* in_sizes, int n_in,
                              void* d_out, int out_size, void* d_ws, size_t ws_size,
                              hipStream_t stream)
{
    const float*         signal = (const float*)d_in[0];
    const unsigned char* mask   = (const unsigned char*)d_in[1];
    const float*         fs     = (const float*)d_in[2];
    const float*         weight = (const float*)d_in[3];
    const float*         bias   = (const float*)d_in[4];

    float* out    = (float*)d_out;
    float* tokens = out;                                   // B*SMAX*E
    float* pm     = tokens + (size_t)B_ * SMAX * E_;       // B*SMAX
    float* wb     = pm + (size_t)B_ * SMAX;                // B
    float* ts     = wb + B_;                               // B*SMAX
    float* fso    = ts + (size_t)B_ * SMAX;                // B

    int4* meta = (int4*)d_ws;

    mt_setup<<<B_, 256, 0, stream>>>(mask, fs, pm, wb, ts, fso, meta);

    dim3 grid(SMAX / 16, B_);
    mt_gemm<<<grid, 256, 0, stream>>>(signal, weight, bias, meta, tokens);
}